// Sentiment_Attention_29652454212397
// MI455X (gfx1250) — compile-verified
//
#include <hip/hip_runtime.h>
#include <stdint.h>

// MI455X (gfx1250) flash-style dual attention.
// Memory-bound: enc_hs (256 MB f32) read exactly ONCE from HBM (~11 us @ 23.3 TB/s floor).
// CDNA5 paths: GLOBAL_LOAD_ASYNC_TO_LDS_B128 (ASYNCcnt) staging with LDS double-buffering,
// V_WMMA_F32_16X16X4_F32 for the P x V context GEMM, wave32 shuffles, 142 KB LDS/block.
// arg/bias terms are constant along the softmax axis -> dropped (softmax invariant).

#define B_ 32
#define S_ 2048
#define H_ 1024
#define TS 16            // tile rows (s-values per LDS tile)
#define TSTRIDE 1040     // padded row stride in words: 1040 % 64 == 16 -> conflict-free
#define RECLEN (2 * H_ + 4)

typedef __attribute__((ext_vector_type(2))) float v2f;
typedef __attribute__((ext_vector_type(8))) float v8f;

__global__ __launch_bounds__(256) void attn_partial_kernel(
    const float* __restrict__ enc, const int* __restrict__ mask,
    const float* __restrict__ w1, const float* __restrict__ w2,
    float* __restrict__ ws, int nchunk, int sc)
{
    __shared__ float tile[2][TS * TSTRIDE];  // 133,120 B double buffer (320 KB LDS/WGP)
    __shared__ float w1s[H_], w2s[H_];       //   8,192 B
    __shared__ float sc1[TS], sc2[TS];
    __shared__ float p1[TS], p2[TS];

    const int t    = threadIdx.x;
    const int lane = t & 31;                 // wave32
    const int wv   = t >> 5;                 // 8 waves
    const int half = lane >> 4;
    const int li   = lane & 15;
    const int b    = blockIdx.y;
    const int chunk = blockIdx.x;
    const int s0   = chunk * sc;
    const int ntiles = sc / TS;

    // ---- async tile prefetch: 16 x b128 per thread, no VGPR staging ----
    // INST_OFFSET is added to BOTH global and LDS addresses; LDS VGPR is pre-biased
    // by r*64 so LDS row stride = 4096 + 64 = 4160 B (padded TSTRIDE).
    auto prefetch = [&](int tl_, int buf_) {
        const float* gsrc = enc + ((size_t)b * S_ + (size_t)(s0 + tl_ * TS)) * H_ + t * 4;
        uint32_t ldsa = (uint32_t)(size_t)(&tile[buf_][0]) + (uint32_t)(t * 16);
#pragma unroll
        for (int r = 0; r < TS; ++r) {
            asm volatile(
                "global_load_async_to_lds_b128 %0, %1, off offset:%2 th:TH_LOAD_NT"
                :: "v"(ldsa + r * 64), "v"(gsrc), "i"(r * 4096) : "memory");
        }
    };

    prefetch(0, 0);  // get HBM stream moving before anything else

    for (int i = t; i < H_; i += 256) { w1s[i] = w1[i]; w2s[i] = w2[i]; }

    // Per-wave WMMA accumulators: wave wv owns h-groups (wv*8+j)*16 .. +15.
    // C row 0 = ctx1, row 1 = ctx2 (A rows >=2 are zero).
    v8f cacc[8];
    {
        v8f z = {};
        for (int j = 0; j < 8; ++j) cacc[j] = z;
    }
    const float NEG_INF = -__builtin_inff();
    float m1 = NEG_INF, m2 = NEG_INF;
    float l1 = 0.f, l2 = 0.f;

    for (int tl = 0; tl < ntiles; ++tl) {
        const int cur = tl & 1;
        asm volatile("s_wait_asynccnt 0x0" ::: "memory");  // my async stores to LDS done
        __syncthreads();  // everyone's async done; prev compute's reads of buf[1-cur] done
        if (tl + 1 < ntiles) prefetch(tl + 1, cur ^ 1);    // overlap next tile with compute
        const float* tb = &tile[cur][0];

        // ---- phase A: scores (16 threads per s-row, stride-16 conflict-free) ----
        {
            const int srow = t >> 4;
            const int part = t & 15;
            float a1 = 0.f, a2 = 0.f;
            for (int j = 0; j < 64; ++j) {
                float e = tb[srow * TSTRIDE + part + 16 * j];
                a1 = fmaf(e, w1s[part + 16 * j], a1);
                a2 = fmaf(e, w2s[part + 16 * j], a2);
            }
            for (int off = 8; off >= 1; off >>= 1) {
                a1 += __shfl_xor(a1, off, 32);   // xor<=8 stays in 16-lane group
                a2 += __shfl_xor(a2, off, 32);
            }
            if (part == 0) {
                const int sg = s0 + tl * TS + srow;
                const bool msk = mask[b * S_ + sg] != 0;   // True = masked
                sc1[srow] = msk ? NEG_INF : a1;
                sc2[srow] = msk ? NEG_INF : a2;
            }
        }
        __syncthreads();

        // ---- phase B: online softmax update (redundant per-thread, deterministic) ----
        float tm1 = NEG_INF, tm2 = NEG_INF;
        for (int i = 0; i < TS; ++i) { tm1 = fmaxf(tm1, sc1[i]); tm2 = fmaxf(tm2, sc2[i]); }
        const float m1n = fmaxf(m1, tm1);
        const float m2n = fmaxf(m2, tm2);
        const float scl1 = (m1 > NEG_INF) ? __expf(m1 - m1n) : 0.f;
        const float scl2 = (m2 > NEG_INF) ? __expf(m2 - m2n) : 0.f;
        if (t < TS)            p1[t]       = (sc1[t] > NEG_INF)      ? __expf(sc1[t] - m1n)      : 0.f;
        else if (t < 2 * TS)   p2[t - TS]  = (sc2[t - TS] > NEG_INF) ? __expf(sc2[t - TS] - m2n) : 0.f;
        m1 = m1n; m2 = m2n;
        __syncthreads();
        if (wv == 0) {   // wave-uniform branch: EXEC stays full per wave
            float su1 = 0.f, su2 = 0.f;
            for (int i = 0; i < TS; ++i) { su1 += p1[i]; su2 += p2[i]; }
            l1 = l1 * scl1 + su1;
            l2 = l2 * scl2 + su2;
        }

        // ---- phase C: ctx += P x tile via WMMA f32 16x16x4 ----
        for (int j = 0; j < 8; ++j) {
            const int hb = (wv * 8 + j) * 16;
            v8f c = cacc[j];
            c[0] *= scl1;            // row 0 = ctx1
            c[1] *= scl2;            // row 1 = ctx2
            for (int kb = 0; kb < TS; kb += 4) {
                const int k0 = kb + 2 * half;
                // A layout (16x4 f32): lane m: v0=A[m][2*half], v1=A[m][2*half+1]
                const float pa0 = p1[k0], pa1 = p1[k0 + 1];
                const float pb0 = p2[k0], pb1 = p2[k0 + 1];
                v2f a;
                a[0] = (li == 0) ? pa0 : ((li == 1) ? pb0 : 0.f);
                a[1] = (li == 0) ? pa1 : ((li == 1) ? pb1 : 0.f);
                // B layout (4x16 f32): lane l: v0=B[2*half][l%16], v1=B[2*half+1][l%16]
                v2f bm;
                bm[0] = tb[(k0 + 0) * TSTRIDE + hb + li];
                bm[1] = tb[(k0 + 1) * TSTRIDE + hb + li];
                c = __builtin_amdgcn_wmma_f32_16x16x4_f32(
                        false, a, false, bm, (short)0, c, false, false);
            }
            cacc[j] = c;
        }
    }
    __syncthreads();

    // ---- write per-chunk partial record: [m1,l1,m2,l2, ctx1[1024], ctx2[1024]] ----
    float* rec = ws + (size_t)(b * nchunk + chunk) * RECLEN;
    if (t == 0) { rec[0] = m1; rec[1] = l1; rec[2] = m2; rec[3] = l2; }
    if (half == 0) {   // C rows 0/1 live in component 0/1 of lanes 0..15
        for (int j = 0; j < 8; ++j) {
            const int hb = (wv * 8 + j) * 16;
            rec[4 + hb + li]      = cacc[j][0];
            rec[4 + H_ + hb + li] = cacc[j][1];
        }
    }
}

__global__ __launch_bounds__(256) void attn_reduce_kernel(
    const float* __restrict__ ws, float* __restrict__ out, int nchunk)
{
    __shared__ float mA1[128], lA1[128], mA2[128], lA2[128], e1[128], e2[128];
    const int t = threadIdx.x;
    const int b = blockIdx.x;
    const float* base = ws + (size_t)b * nchunk * RECLEN;
    const float NEG_INF = -__builtin_inff();

    if (t < nchunk) {
        const float* r = base + (size_t)t * RECLEN;
        mA1[t] = r[0]; lA1[t] = r[1]; mA2[t] = r[2]; lA2[t] = r[3];
    }
    __syncthreads();
    float M1 = NEG_INF, M2 = NEG_INF;
    for (int c = 0; c < nchunk; ++c) { M1 = fmaxf(M1, mA1[c]); M2 = fmaxf(M2, mA2[c]); }
    if (t < nchunk) {
        e1[t] = (mA1[t] > NEG_INF) ? __expf(mA1[t] - M1) : 0.f;
        e2[t] = (mA2[t] > NEG_INF) ? __expf(mA2[t] - M2) : 0.f;
    }
    __syncthreads();
    float L1 = 0.f, L2 = 0.f;
    for (int c = 0; c < nchunk; ++c) { L1 += lA1[c] * e1[c]; L2 += lA2[c] * e2[c]; }
    const float inv1 = 1.0f / L1, inv2 = 1.0f / L2;

    for (int k = 0; k < H_ / 256; ++k) {
        const int h = t + k * 256;
        float acc1 = 0.f, acc2 = 0.f;
        for (int c = 0; c < nchunk; ++c) {
            const float* r = base + (size_t)c * RECLEN;
            acc1 = fmaf(r[4 + h],      e1[c], acc1);
            acc2 = fmaf(r[4 + H_ + h], e2[c], acc2);
        }
        out[(size_t)b * (2 * H_) + h]      = acc1 * inv1;
        out[(size_t)b * (2 * H_) + H_ + h] = acc2 * inv2;
    }
}

extern "C" void kernel_launch(void* const* d_in, const int* in_sizes, int n_in,
                              void* d_out, int out_size, void* d_ws, size_t ws_size,
                              hipStream_t stream) {
    // setup_inputs order: arg1, arg2, enc_hs, src_mask, w1_w, w1_b, w2_w, w2_b.
    // arg/bias terms are softmax-invariant -> unused. src_mask (bool) assumed
    // shipped as int32 per the harness's "integer -> const int*" convention.
    const float* enc  = (const float*)d_in[2];
    const int*   mask = (const int*)d_in[3];
    const float* w1   = (const float*)d_in[4];
    const float* w2   = (const float*)d_in[6];
    float* out = (float*)d_out;
    float* ws  = (float*)d_ws;

    int nchunk = 64;   // 2048 blocks target; shrink to fit workspace
    while (nchunk > 1 && (size_t)B_ * nchunk * RECLEN * sizeof(float) > ws_size)
        nchunk >>= 1;
    const int sc = S_ / nchunk;

    attn_partial_kernel<<<dim3(nchunk, B_), 256, 0, stream>>>(enc, mask, w1, w2, ws, nchunk, sc);
    attn_reduce_kernel<<<B_, 256, 0, stream>>>(ws, out, nchunk);
}